// snn_model_20950850470254
// MI455X (gfx1250) — compile-verified
//
#include <hip/hip_runtime.h>

typedef __attribute__((ext_vector_type(2))) float v2f;
typedef __attribute__((ext_vector_type(8))) float v8f;

#define TSTEPS 20
#define MPAD   32
#define DECAY  0.9375f
#define THRES  1.0f

// ---------------------------------------------------------------------------
// Kernel 1: input-layer integrate & fire (no decay). Builds S0 (MPAD x 3072),
// rows 20..31 zeroed (padding for the 32-row WMMA GEMM).
// ---------------------------------------------------------------------------
__global__ void input_spikes_kernel(const float* __restrict__ image,
                                    float* __restrict__ S0, int n) {
    int i = blockIdx.x * blockDim.x + threadIdx.x;
    if (i >= n) return;
    float x = image[i];
    float mi = 0.0f;
    for (int t = 0; t < TSTEPS; ++t) {
        mi += x;
        float s = (mi >= THRES) ? 1.0f : 0.0f;
        mi -= THRES * s;
        S0[(size_t)t * n + i] = s;
    }
    for (int t = TSTEPS; t < MPAD; ++t) S0[(size_t)t * n + i] = 0.0f;
}

// ---------------------------------------------------------------------------
// Z = S (MPAD x K, rows>=20 zero) @ W (K x LDW), fp32 WMMA 16x16x4.
// All geometry is compile-time constant so every access in the K-loop is a
// single vector load with an immediate offset off a pointer bumped once per
// 32-K chunk (max offset 31*LDW*4 ~= 1MB < 24-bit signed IOFFSET).
//
// Block = 256 threads = 8 waves; wave -> (m_tile in {0,1}, n_tile in {0..3});
// block covers 64 columns; grid.x = LDW/64. Each weight element is read by
// two waves (m_tile 0/1) -> W streamed from HBM exactly once (the roofline floor:
// ~369 MB total / 23.3 TB/s ~= 16 us). Weight loads are non-temporal so the
// 369 MB stream does not evict the ~3 MB spike/Z working set from L2.
//
// ISA operand layouts (32-bit, wave32):
//   A 16x4 : vgpr j, lane l : A[M=l&15][K = 2*(l>>4) + j]
//   B 4x16 : vgpr j, lane l : B[K = 2*(l>>4) + j][N = l&15]
//   C 16x16: vgpr r, lane l : C[M = 8*(l>>4) + r][N = l&15]
// ---------------------------------------------------------------------------
template <int K, int LDS_, int LDW>
__global__ void spike_gemm_wmma(const float* __restrict__ S,
                                const float* __restrict__ W,
                                float* __restrict__ Z) {
    const int tid    = threadIdx.x;
    const int wave   = tid >> 5;
    const int lane   = tid & 31;
    const int half   = lane >> 4;    // 0 or 1
    const int l16    = lane & 15;
    const int m_tile = wave & 1;     // rows 0..15 or 16..31
    const int n_tile = wave >> 1;    // 4 column tiles per block
    const int n0     = blockIdx.x * 64 + n_tile * 16;

    const int arow = m_tile * 16 + l16;
    // pre-offset by the K-sub-index owned by this half-wave
    const float* __restrict__ Sp = S + arow * LDS_ + 2 * half;
    const float* __restrict__ Bp = W + (2 * half) * LDW + n0 + l16;

    v8f acc = {};
    for (int k = 0; k < K; k += 32) {
        // prefetch the weight stream 64 K-rows ahead (uniform branch; the
        // 64*LDW*4 = 2MB displacement is a compile-time immediate)
        if (k + 64 < K)
            __builtin_prefetch(Bp + 64 * LDW, 0, 0);
#pragma unroll
        for (int kk = 0; kk < 32; kk += 4) {
            v2f a, b;
            a.x = Sp[kk];
            a.y = Sp[kk + 1];
            b.x = __builtin_nontemporal_load(Bp + kk * LDW);
            b.y = __builtin_nontemporal_load(Bp + (kk + 1) * LDW);
            acc = __builtin_amdgcn_wmma_f32_16x16x4_f32(
                false, a, false, b, (short)0, acc, false, false);
        }
        Sp += 32;
        Bp += 32 * LDW;
    }

#pragma unroll
    for (int r = 0; r < 8; ++r) {
        int t = m_tile * 16 + half * 8 + r;   // output timestep row
        if (t < TSTEPS)
            Z[t * LDW + n0 + l16] = acc[r];
    }
}

// ---------------------------------------------------------------------------
// Per-neuron LIF scan over the 20 timesteps. Consumes Z (rows 0..19) and
// produces the SHIFTED spike matrix (row 0 = zeros, row t = spikes of step
// t-1), rows 20..31 zero-padded for the next WMMA GEMM.
// ---------------------------------------------------------------------------
__global__ void lif_shift_kernel(const float* __restrict__ Z,
                                 float* __restrict__ Sshift, int N) {
    int j = blockIdx.x * blockDim.x + threadIdx.x;
    if (j >= N) return;
    float m = 0.0f;
    Sshift[j] = 0.0f;                       // row 0 (pipeline delay)
    for (int t = 0; t < TSTEPS; ++t) {
        float v = m + Z[(size_t)t * N + j];
        float fire = (v >= THRES) ? 1.0f : 0.0f;
        v -= THRES * fire;
        m = (fire > 0.0f) ? v : v * DECAY;
        if (t < TSTEPS - 1)
            Sshift[(size_t)(t + 1) * N + j] = fire;
    }
    for (int t = TSTEPS; t < MPAD; ++t) Sshift[(size_t)t * N + j] = 0.0f;
}

// ---------------------------------------------------------------------------
// Z3[t][j] = dot(S2shift[t][:], W3[:][j]); N=10 so block reduction beats
// WMMA here (W3 is only 320 KB total). grid = (10, 20), block = 256.
// ---------------------------------------------------------------------------
__global__ void layer3_gemv_kernel(const float* __restrict__ S2,
                                   const float* __restrict__ W3,
                                   float* __restrict__ Z3, int K) {
    __shared__ float red[256];
    const int j = blockIdx.x;   // output neuron 0..9
    const int t = blockIdx.y;   // timestep 0..19
    const int tid = threadIdx.x;
    const float* __restrict__ s = S2 + (size_t)t * K;
    float sum = 0.0f;
    for (int i = tid; i < K; i += 256)
        sum += s[i] * W3[(size_t)i * 10 + j];
    red[tid] = sum;
    __syncthreads();
    for (int off = 128; off > 0; off >>= 1) {
        if (tid < off) red[tid] += red[tid + off];
        __syncthreads();
    }
    if (tid == 0) Z3[t * 10 + j] = red[0];
}

// ---------------------------------------------------------------------------
// Final LIF over the 10 output neurons + write spikes_out (21 x 10, row 0
// zeros).
// ---------------------------------------------------------------------------
__global__ void output_kernel(const float* __restrict__ Z3,
                              float* __restrict__ out) {
    int j = threadIdx.x;
    if (j >= 10) return;
    out[j] = 0.0f;
    float m = 0.0f;
    for (int t = 0; t < TSTEPS; ++t) {
        float v = m + Z3[t * 10 + j];
        float fire = (v >= THRES) ? 1.0f : 0.0f;
        v -= THRES * fire;
        m = (fire > 0.0f) ? v : v * DECAY;
        out[(t + 1) * 10 + j] = fire;
    }
}

// ---------------------------------------------------------------------------
extern "C" void kernel_launch(void* const* d_in, const int* in_sizes, int n_in,
                              void* d_out, int out_size, void* d_ws, size_t ws_size,
                              hipStream_t stream) {
    const float* image = (const float*)d_in[0];   // 3072
    const float* W1    = (const float*)d_in[1];   // 3072 x 8192
    const float* W2    = (const float*)d_in[2];   // 8192 x 8192
    const float* W3    = (const float*)d_in[3];   // 8192 x 10
    float* out = (float*)d_out;                   // 21 x 10

    const int N0 = 3072, N1 = 8192, N2 = 8192;

    // workspace layout (floats), ~3.5 MB total
    float* ws  = (float*)d_ws;
    float* S0  = ws;                          // MPAD x 3072
    float* S1  = S0 + (size_t)MPAD * N0;      // MPAD x 8192 (shifted spikes L1)
    float* S2  = S1 + (size_t)MPAD * N1;      // MPAD x 8192 (shifted spikes L2)
    float* Zb  = S2 + (size_t)MPAD * N2;      // MPAD x 8192 (Z1 then Z2, reused)
    float* Z3  = Zb + (size_t)MPAD * N2;      // 20 x 10 (+pad)

    // 1) input spikes for all 20 steps (depends only on image)
    input_spikes_kernel<<<(N0 + 255) / 256, 256, 0, stream>>>(image, S0, N0);

    // 2) Z1 = S0 @ W1   (reads W1 exactly once)
    spike_gemm_wmma<3072, 3072, 8192><<<N1 / 64, 256, 0, stream>>>(S0, W1, Zb);

    // 3) layer-1 LIF scan -> shifted spike matrix S1
    lif_shift_kernel<<<N1 / 256, 256, 0, stream>>>(Zb, S1, N1);

    // 4) Z2 = S1 @ W2   (reads W2 exactly once)
    spike_gemm_wmma<8192, 8192, 8192><<<N2 / 64, 256, 0, stream>>>(S1, W2, Zb);

    // 5) layer-2 LIF scan -> shifted spike matrix S2
    lif_shift_kernel<<<N2 / 256, 256, 0, stream>>>(Zb, S2, N2);

    // 6) Z3 = S2 @ W3 (N=10 -> block reductions)
    layer3_gemv_kernel<<<dim3(10, TSTEPS), 256, 0, stream>>>(S2, W3, Z3, N2);

    // 7) output LIF + spike matrix (row 0 zeros)
    output_kernel<<<1, 32, 0, stream>>>(Z3, out);
}